// MACE_32590211842659
// MI455X (gfx1250) — compile-verified
//
#include <hip/hip_runtime.h>
#include <math.h>

#define C_CH 64
#define E_CHUNK 16384

typedef __attribute__((ext_vector_type(2))) float v2f;
typedef __attribute__((ext_vector_type(8))) float v8f;

// PATHS[1] enumeration order (first 3 entries == PATHS[0])
__constant__ int cL1[13] = {0,0,0,1,1,1,1,1,2,2,2,2,2};
__constant__ int cL2[13] = {0,1,2,0,1,1,2,3,0,1,2,2,3};
__constant__ int cL3[13] = {0,1,2,1,0,2,1,2,2,1,0,2,1};
__constant__ int cYoff[4] = {0,1,4,9};   // offset of Y_l within 16-comp vector
__constant__ int cKoff[3] = {0,1,4};     // offset of l3 block within 9-comp accumulator

// ---------------- CG coefficient generation (device, double precision) --------
__device__ double dfact(int n){ double r=1.0; for(int i=2;i<=n;++i) r*=(double)i; return r; }

__device__ double cg_coef(int j1,int m1,int j2,int m2,int j3,int m3){
  if (m1+m2 != m3) return 0.0;
  double pref = sqrt((2.0*j3+1.0)*dfact(j3+j1-j2)*dfact(j3-j1+j2)*dfact(j1+j2-j3)/dfact(j1+j2+j3+1));
  pref *= sqrt(dfact(j3+m3)*dfact(j3-m3)*dfact(j1-m1)*dfact(j1+m1)*dfact(j2-m2)*dfact(j2+m2));
  double s = 0.0;
  for (int k=0;k<=j1+j2-j3;++k){
    int d1=j1+j2-j3-k, d2=j1-m1-k, d3=j2+m2-k, d4=j3-j2+m1+k, d5=j3-j1-m2+k;
    if (d1<0||d2<0||d3<0||d4<0||d5<0) continue;
    double den = dfact(k)*dfact(d1)*dfact(d2)*dfact(d3)*dfact(d4)*dfact(d5);
    s += ((k&1)?-1.0:1.0)/den;
  }
  return pref*s;
}

__device__ void real_basis7(int l, double Ar[7][7], double Ai[7][7]){
  for (int i=0;i<7;++i) for (int j=0;j<7;++j){ Ar[i][j]=0.0; Ai[i][j]=0.0; }
  const double is2 = 0.70710678118654752440;
  for (int mp=-l; mp<=l; ++mp){
    if (mp>0){
      Ar[l+mp][l+mp] = ((mp&1)?-1.0:1.0)*is2;
      Ar[l+mp][l-mp] = is2;
    } else if (mp==0){
      Ar[l][l] = 1.0;
    } else {
      int m = -mp;
      Ai[l+mp][l-m] = is2;
      Ai[l+mp][l+m] = -(((m&1)?-1.0:1.0))*is2;
    }
  }
}

__global__ void cg_init_kernel(float* __restrict__ cg){
  int p = threadIdx.x;
  if (p >= 13) return;
  int l1=cL1[p], l2=cL2[p], l3=cL3[p];
  double A1r[7][7],A1i[7][7],A2r[7][7],A2i[7][7],A3r[7][7],A3i[7][7];
  real_basis7(l1,A1r,A1i);
  real_basis7(l2,A2r,A2i);
  real_basis7(l3,A3r,A3i);
  double Cre[5][7][5], Cim[5][7][5];
  for (int i=0;i<5;++i)for(int j=0;j<7;++j)for(int k=0;k<5;++k){Cre[i][j][k]=0.0;Cim[i][j][k]=0.0;}
  for (int i=0;i<=2*l1;++i)
   for (int j=0;j<=2*l2;++j)
    for (int k=0;k<=2*l3;++k){
      double cr=0.0, ci=0.0;
      for (int a=0;a<=2*l1;++a){
        double a1r=A1r[i][a], a1i=A1i[i][a];
        if (a1r==0.0 && a1i==0.0) continue;
        for (int b=0;b<=2*l2;++b){
          double a2r=A2r[j][b], a2i=A2i[j][b];
          if (a2r==0.0 && a2i==0.0) continue;
          int cm = (a-l1)+(b-l2);
          if (cm < -l3 || cm > l3) continue;
          double cc = cg_coef(l1,a-l1,l2,b-l2,l3,cm);
          if (cc == 0.0) continue;
          double tr = a1r*a2r - a1i*a2i;
          double ti = a1r*a2i + a1i*a2r;
          double a3r = A3r[k][l3+cm];
          double a3i = -A3i[k][l3+cm];   // conjugate
          cr += (tr*a3r - ti*a3i)*cc;
          ci += (tr*a3i + ti*a3r)*cc;
        }
      }
      Cre[i][j][k]=cr; Cim[i][j][k]=ci;
    }
  double sre=0.0, sim=0.0;
  for (int i=0;i<=2*l1;++i)for(int j=0;j<=2*l2;++j)for(int k=0;k<=2*l3;++k){
    sre += fabs(Cre[i][j][k]); sim += fabs(Cim[i][j][k]);
  }
  bool useRe = (sre >= sim);
  double nrm = 0.0;
  for (int i=0;i<=2*l1;++i)for(int j=0;j<=2*l2;++j)for(int k=0;k<=2*l3;++k){
    double v = useRe ? Cre[i][j][k] : Cim[i][j][k];
    nrm += v*v;
  }
  nrm = sqrt(nrm);
  if (nrm < 1e-12) nrm = 1e-12;
  double invn = 1.0/nrm;
  float* dst = cg + p*175;    // slot layout [5][7][5]
  for (int i=0;i<5;++i)for(int j=0;j<7;++j)for(int k=0;k<5;++k){
    double v = 0.0;
    if (i<=2*l1 && j<=2*l2 && k<=2*l3) v = (useRe?Cre[i][j][k]:Cim[i][j][k])*invn;
    dst[(i*7+j)*5+k] = (float)v;
  }
}

// ---------------- edge geometry: Y (16) + radial basis (8) --------------------
__global__ void geom_kernel(const float* __restrict__ pos,
                            const int* __restrict__ snd, const int* __restrict__ rcv,
                            float* __restrict__ Y, float* __restrict__ rad, int E)
{
  int e = blockIdx.x*blockDim.x + threadIdx.x;
  if (e >= E) return;
  int s = snd[e], t = rcv[e];
  float rx = pos[t*3+0]-pos[s*3+0];
  float ry = pos[t*3+1]-pos[s*3+1];
  float rz = pos[t*3+2]-pos[s*3+2];
  float r = sqrtf(rx*rx+ry*ry+rz*rz + 1e-12f);
  float inv = 1.0f/r;
  float x=rx*inv, y=ry*inv, z=rz*inv;
  float* Ye = Y + (size_t)e*16;
  const float s3  = 1.7320508075688772f;   // sqrt(3)
  const float s15 = 3.872983346207417f;    // sqrt(15)
  const float s5h = 1.118033988749895f;    // sqrt(5)/2
  Ye[0]=1.f;
  Ye[1]=s3*y; Ye[2]=s3*z; Ye[3]=s3*x;
  Ye[4]=s15*x*y;
  Ye[5]=s15*y*z;
  Ye[6]=s5h*(3.f*z*z-1.f);
  Ye[7]=s15*x*z;
  Ye[8]=0.5f*s15*(x*x-y*y);
  const float c30 = 2.091650066335189f;    // sqrt(35/8)
  const float c31 = 10.246950765959598f;   // sqrt(105)
  const float c32 = 1.6201851746019651f;   // sqrt(21/8)
  const float c33 = 1.3228756555322954f;   // sqrt(7)/2
  float zz5 = 5.f*z*z;
  Ye[9]  = c30*y*(3.f*x*x - y*y);
  Ye[10] = c31*x*y*z;
  Ye[11] = c32*y*(zz5-1.f);
  Ye[12] = c33*z*(zz5-3.f);
  Ye[13] = c32*x*(zz5-1.f);
  Ye[14] = 0.5f*c31*z*(x*x-y*y);
  Ye[15] = c30*x*(x*x-3.f*y*y);
  // radial: bessel * envelope
  float rs = fmaxf(r, 1e-6f);
  float tt = 2.f*(1.f - r*0.2f);
  float env = (tt > 0.f) ? (1.2f*expf(0.5f))*expf(-1.f/tt) : 0.f;
  const float pref = 0.6324555320336759f;  // sqrt(2/5)
  float* Re = rad + (size_t)e*8;
  float arg = 3.14159265358979323846f * rs * 0.2f;
  #pragma unroll
  for (int n=1;n<=8;++n) Re[n-1] = pref * sinf((float)n*arg) / rs * env;
}

// ---------------- node embedding ----------------------------------------------
__global__ void embed_kernel(const int* __restrict__ species, const float* __restrict__ wemb,
                             float* __restrict__ h, int N){
  int i = blockIdx.x*blockDim.x + threadIdx.x;
  if (i >= N*C_CH) return;
  int n = i >> 6, c = i & 63;
  h[i] = wemb[species[n]*C_CH + c];
}

__global__ void zero_kernel(float* __restrict__ p, int n){
  int i = blockIdx.x*blockDim.x + threadIdx.x;
  if (i < n) p[i] = 0.f;
}

// ---------------- fp32 WMMA GEMM: C[M,N] = act(A[M,K] @ B[K,N]) ---------------
// One wave computes a 16x64 output tile (4 accumulators) using
// V_WMMA_F32_16X16X4_F32; one A fragment feeds 4 WMMAs per k-step.
// A fragment (16x4 f32): lanes 0-15 -> M=lane, K=0/1; lanes 16-31 -> M=lane-16, K=2/3.
// B fragment (4x16 f32): lanes 0-15 -> K rows 0/1; lanes 16-31 -> K rows 2/3; col=lane&15.
// C/D fragment: VGPR r -> row r (lanes<16) / row r+8 (lanes>=16); col=lane&15.
// Requires N % 64 == 0 (true for all uses: 64, 192, 832).
__global__ void gemm_wmma(const float* __restrict__ A, const float* __restrict__ B,
                          float* __restrict__ Cp, int M, int N, int K,
                          int act, int accum)
{
  const int tileM = blockIdx.x << 4;
  const int tileN = blockIdx.y << 6;
  const int lane  = threadIdx.x & 31;
  const int mloc  = lane & 15;
  const int kh    = (lane >> 4) << 1;       // 0 or 2
  int arow = tileM + mloc; if (arow >= M) arow = M - 1;
  const float* pA = A + (size_t)arow * K + kh;
  const float* pB = B + (size_t)kh * N + tileN + mloc;
  const size_t bStep = (size_t)4 * N;
  __builtin_prefetch(pA, 0, 0);             // global_prefetch_b8: pull A row
  v8f acc0 = {0.f,0.f,0.f,0.f,0.f,0.f,0.f,0.f};
  v8f acc1 = acc0, acc2 = acc0, acc3 = acc0;
  for (int kk = 0; kk < K; kk += 4) {
    v2f a, b0, b1, b2, b3;
    a.x  = pA[0];      a.y  = pA[1];
    b0.x = pB[0];      b0.y = pB[N];
    b1.x = pB[16];     b1.y = pB[16 + N];
    b2.x = pB[32];     b2.y = pB[32 + N];
    b3.x = pB[48];     b3.y = pB[48 + N];
    acc0 = __builtin_amdgcn_wmma_f32_16x16x4_f32(false, a, false, b0, (short)0, acc0, false, false);
    acc1 = __builtin_amdgcn_wmma_f32_16x16x4_f32(false, a, false, b1, (short)0, acc1, false, false);
    acc2 = __builtin_amdgcn_wmma_f32_16x16x4_f32(false, a, false, b2, (short)0, acc2, false, false);
    acc3 = __builtin_amdgcn_wmma_f32_16x16x4_f32(false, a, false, b3, (short)0, acc3, false, false);
    pA += 4;
    pB += bStep;
  }
  const int rbase = (lane >> 4) << 3;
  v8f accs[4] = {acc0, acc1, acc2, acc3};
  #pragma unroll
  for (int t = 0; t < 4; ++t) {
    const int col = tileN + (t << 4) + mloc;
    #pragma unroll
    for (int r = 0; r < 8; ++r) {
      int row = tileM + rbase + r;
      if (row < M) {
        size_t idx = (size_t)row * N + col;
        float v = accs[t][r];
        if (accum) v += Cp[idx];
        if (act == 1) v = v / (1.f + __expf(-v));   // silu
        Cp[idx] = v;
      }
    }
  }
}

// ---------------- per-edge CG tensor product + scatter-add --------------------
// One 64-thread block per edge; thread = channel c. Node feature layout [n][k][c].
__global__ void edge_tp(int eBase, int nPaths,
                        const int* __restrict__ senders, const int* __restrict__ receivers,
                        const float* __restrict__ Y, const float* __restrict__ wChunk,
                        const float* __restrict__ h0, const float* __restrict__ h1,
                        const float* __restrict__ h2,
                        const float* __restrict__ cg,
                        float* __restrict__ msg0, float* __restrict__ msg1,
                        float* __restrict__ msg2)
{
  const int e = eBase + blockIdx.x;
  const int c = threadIdx.x;
  __shared__ float sY[16];
  if (c < 16) sY[c] = Y[(size_t)e*16 + c];
  __syncthreads();
  const int snd = senders[e];
  const int rcv = receivers[e];
  const float* wrow = wChunk + (size_t)blockIdx.x * nPaths * C_CH + c;
  float acc[9];
  #pragma unroll
  for (int q=0;q<9;++q) acc[q]=0.f;
  for (int p=0;p<nPaths;++p){
    const int l1=cL1[p], l2=cL2[p], l3=cL3[p];
    const int K1=2*l1+1, K2=2*l2+1, K3=2*l3+1;
    const float* hsrc = (l1==0)?h0:((l1==1)?h1:h2);
    float hv[5];
    for (int i=0;i<K1;++i) hv[i] = hsrc[((size_t)snd*K1 + i)*C_CH + c];
    const float* cgp = cg + p*175;
    const float* Yl  = sY + cYoff[l2];
    float tmp[5] = {0.f,0.f,0.f,0.f,0.f};
    for (int i=0;i<K1;++i){
      const float hvi = hv[i];
      for (int j=0;j<K2;++j){
        const float hy = hvi * Yl[j];
        const float* cgr = cgp + (i*7+j)*5;
        for (int k=0;k<K3;++k) tmp[k] = fmaf(hy, cgr[k], tmp[k]);
      }
    }
    const float wp = wrow[p*C_CH];
    const int ko = cKoff[l3];
    for (int k=0;k<K3;++k) acc[ko+k] = fmaf(wp, tmp[k], acc[ko+k]);
  }
  const float s = 1.0f/16.0f;   // 1 / AVG_NEI
  atomicAdd(&msg0[(size_t)rcv*C_CH + c], acc[0]*s);
  #pragma unroll
  for (int k=0;k<3;++k) atomicAdd(&msg1[((size_t)rcv*3 + k)*C_CH + c], acc[1+k]*s);
  #pragma unroll
  for (int k=0;k<5;++k) atomicAdd(&msg2[((size_t)rcv*5 + k)*C_CH + c], acc[4+k]*s);
}

// ---------------- readout pack: [n][k][d] -> out[n, colBase + d*K + k] --------
__global__ void pack_kernel(const float* __restrict__ t, float* __restrict__ out,
                            int N, int Kd, int colBase){
  int idx = blockIdx.x*blockDim.x + threadIdx.x;
  int total = N*C_CH*Kd;
  if (idx >= total) return;
  int n = idx / (C_CH*Kd);
  int rem = idx % (C_CH*Kd);
  int d = rem / Kd;
  int k = rem % Kd;
  out[(size_t)n*1152 + colBase + d*Kd + k] = t[((size_t)n*Kd + k)*C_CH + d];
}

// ------------------------------------------------------------------------------
extern "C" void kernel_launch(void* const* d_in, const int* in_sizes, int n_in,
                              void* d_out, int out_size, void* d_ws, size_t ws_size,
                              hipStream_t stream)
{
  (void)n_in; (void)out_size; (void)ws_size;
  const float* positions = (const float*)d_in[0];
  const int*   species   = (const int*)d_in[1];
  const int*   senders   = (const int*)d_in[2];
  const int*   receivers = (const int*)d_in[3];
  const float* w_embed   = (const float*)d_in[4];
  const float* rw1_0     = (const float*)d_in[5];
  const float* rw2_0     = (const float*)d_in[6];
  const float* mix0      = (const float*)d_in[7];
  const float* read0     = (const float*)d_in[8];
  const float* rw1_1     = (const float*)d_in[9];
  const float* rw2_1     = (const float*)d_in[10];
  const float* mix1      = (const float*)d_in[11];
  const float* skip1     = (const float*)d_in[12];
  const float* read1     = (const float*)d_in[13];
  float* out = (float*)d_out;

  const int N = in_sizes[1];   // 10000
  const int E = in_sizes[2];   // 160000

  float* base = (float*)d_ws;
  size_t off = 0;
  auto alloc = [&](size_t n)->float*{ float* p = base + off; off += (n + 63) & ~(size_t)63; return p; };

  float* cgbuf = alloc(13*175);
  float* Yb    = alloc((size_t)E*16);
  float* radb  = alloc((size_t)E*8);
  float* hemb  = alloc((size_t)N*C_CH);
  float* msg0  = alloc((size_t)N*C_CH);
  float* msg1  = alloc((size_t)N*3*C_CH);
  float* msg2  = alloc((size_t)N*5*C_CH);
  float* hA0   = alloc((size_t)N*C_CH);
  float* hA1   = alloc((size_t)N*3*C_CH);
  float* hA2   = alloc((size_t)N*5*C_CH);
  float* hB0   = alloc((size_t)N*C_CH);
  float* hB1   = alloc((size_t)N*3*C_CH);
  float* hB2   = alloc((size_t)N*5*C_CH);
  float* tmp   = alloc((size_t)N*5*C_CH);
  float* hid   = alloc((size_t)E_CHUNK*C_CH);
  float* wbuf  = alloc((size_t)E_CHUNK*13*C_CH);

  cg_init_kernel<<<1, 16, 0, stream>>>(cgbuf);
  geom_kernel<<<(E+255)/256, 256, 0, stream>>>(positions, senders, receivers, Yb, radb, E);
  embed_kernel<<<(N*C_CH+255)/256, 256, 0, stream>>>(species, w_embed, hemb, N);

  // ---------------- layer 0 ----------------
  zero_kernel<<<(N*C_CH+255)/256,256,0,stream>>>(msg0, N*C_CH);
  zero_kernel<<<(N*3*C_CH+255)/256,256,0,stream>>>(msg1, N*3*C_CH);
  zero_kernel<<<(N*5*C_CH+255)/256,256,0,stream>>>(msg2, N*5*C_CH);

  for (int eb = 0; eb < E; eb += E_CHUNK) {
    int cnt = E - eb; if (cnt > E_CHUNK) cnt = E_CHUNK;
    gemm_wmma<<<dim3(cnt/16, 1), 32, 0, stream>>>(radb + (size_t)eb*8, rw1_0, hid, cnt, 64, 8, 1, 0);
    gemm_wmma<<<dim3(cnt/16, 3), 32, 0, stream>>>(hid, rw2_0, wbuf, cnt, 192, 64, 0, 0);
    edge_tp<<<cnt, 64, 0, stream>>>(eb, 3, senders, receivers, Yb, wbuf,
                                    hemb, hemb, hemb, cgbuf, msg0, msg1, msg2);
  }
  gemm_wmma<<<dim3(N/16,1),   32,0,stream>>>(msg0, mix0,        hA0, N,   64, 64, 0, 0);
  gemm_wmma<<<dim3(N*3/16,1), 32,0,stream>>>(msg1, mix0+4096,   hA1, N*3, 64, 64, 0, 0);
  gemm_wmma<<<dim3(N*5/16,1), 32,0,stream>>>(msg2, mix0+8192,   hA2, N*5, 64, 64, 0, 0);

  {
    const float* hAs[3] = {hA0,hA1,hA2};
    const int lOff[3] = {0,64,256};
    for (int l=0;l<3;++l){
      int Kd = 2*l+1;
      gemm_wmma<<<dim3(N*Kd/16,1),32,0,stream>>>(hAs[l], read0 + l*4096, tmp, N*Kd, 64, 64, 0, 0);
      pack_kernel<<<(N*C_CH*Kd+255)/256,256,0,stream>>>(tmp, out, N, Kd, lOff[l]);
    }
  }

  // ---------------- layer 1 ----------------
  zero_kernel<<<(N*C_CH+255)/256,256,0,stream>>>(msg0, N*C_CH);
  zero_kernel<<<(N*3*C_CH+255)/256,256,0,stream>>>(msg1, N*3*C_CH);
  zero_kernel<<<(N*5*C_CH+255)/256,256,0,stream>>>(msg2, N*5*C_CH);

  for (int eb = 0; eb < E; eb += E_CHUNK) {
    int cnt = E - eb; if (cnt > E_CHUNK) cnt = E_CHUNK;
    gemm_wmma<<<dim3(cnt/16, 1),  32, 0, stream>>>(radb + (size_t)eb*8, rw1_1, hid, cnt, 64, 8, 1, 0);
    gemm_wmma<<<dim3(cnt/16, 13), 32, 0, stream>>>(hid, rw2_1, wbuf, cnt, 832, 64, 0, 0);
    edge_tp<<<cnt, 64, 0, stream>>>(eb, 13, senders, receivers, Yb, wbuf,
                                    hA0, hA1, hA2, cgbuf, msg0, msg1, msg2);
  }
  // hB = msg @ mix1 ; hB += hA @ skip1
  gemm_wmma<<<dim3(N/16,1),   32,0,stream>>>(msg0, mix1,       hB0, N,   64, 64, 0, 0);
  gemm_wmma<<<dim3(N*3/16,1), 32,0,stream>>>(msg1, mix1+4096,  hB1, N*3, 64, 64, 0, 0);
  gemm_wmma<<<dim3(N*5/16,1), 32,0,stream>>>(msg2, mix1+8192,  hB2, N*5, 64, 64, 0, 0);
  gemm_wmma<<<dim3(N/16,1),   32,0,stream>>>(hA0,  skip1,      hB0, N,   64, 64, 0, 1);
  gemm_wmma<<<dim3(N*3/16,1), 32,0,stream>>>(hA1,  skip1+4096, hB1, N*3, 64, 64, 0, 1);
  gemm_wmma<<<dim3(N*5/16,1), 32,0,stream>>>(hA2,  skip1+8192, hB2, N*5, 64, 64, 0, 1);

  {
    const float* hBs[3] = {hB0,hB1,hB2};
    const int lOff[3] = {0,64,256};
    for (int l=0;l<3;++l){
      int Kd = 2*l+1;
      gemm_wmma<<<dim3(N*Kd/16,1),32,0,stream>>>(hBs[l], read1 + l*4096, tmp, N*Kd, 64, 64, 0, 0);
      pack_kernel<<<(N*C_CH*Kd+255)/256,256,0,stream>>>(tmp, out, N, Kd, 576 + lOff[l]);
    }
  }
}